// FilterDetections_91147795956366
// MI455X (gfx1250) — compile-verified
//
#include <hip/hip_runtime.h>
#include <stdint.h>
#include <stddef.h>

#ifndef __has_builtin
#define __has_builtin(x) 0
#endif

#define BATCH   8
#define NBOX    50000
#define NCLS    80
#define KCAND   500
#define MAXDET  300
#define TILE    4096
#define NTILES  ((NBOX + TILE - 1) / TILE)
#define MAXC    16384
#define SCORE_T 0.05f
#define NMS_TH  0.5f

#if defined(__HIP_DEVICE_COMPILE__) && __has_builtin(__builtin_amdgcn_tensor_load_to_lds) && __has_builtin(__builtin_amdgcn_s_wait_tensorcnt)
#define HAS_TDM 1
#else
#define HAS_TDM 0
#endif

// ---------------- helpers ----------------

__device__ __forceinline__ unsigned ord_bits(unsigned u) {
  // monotonic float -> uint mapping (handles -inf correctly)
  return u ^ ((u & 0x80000000u) ? 0xFFFFFFFFu : 0x80000000u);
}
__device__ __forceinline__ float from_ord(unsigned ou) {
  unsigned u = (ou & 0x80000000u) ? (ou ^ 0x80000000u) : ~ou;
  return __uint_as_float(u);
}
__device__ __forceinline__ unsigned long long key64_of(unsigned scorebits, unsigned idx) {
  // descending score, ascending index (stable top_k tie-break); keys are unique
  return ((unsigned long long)ord_bits(scorebits) << 32) | (unsigned)(~idx);
}

__device__ __forceinline__ void bitonic_desc_512(unsigned long long* key, int tid, int nt) {
  for (unsigned k = 2; k <= 512; k <<= 1) {
    for (unsigned j = k >> 1; j > 0; j >>= 1) {
      __syncthreads();
      for (unsigned i = (unsigned)tid; i < 512; i += (unsigned)nt) {
        unsigned ixj = i ^ j;
        if (ixj > i) {
          bool up = ((i & k) == 0);
          unsigned long long a = key[i], bb = key[ixj];
          bool sw = up ? (a < bb) : (a > bb);   // descending overall
          if (sw) { key[i] = bb; key[ixj] = a; }
        }
      }
    }
  }
  __syncthreads();
}

#if HAS_TDM
typedef unsigned int u32x4 __attribute__((ext_vector_type(4)));
typedef int          i32x4 __attribute__((ext_vector_type(4)));
typedef int          i32x8 __attribute__((ext_vector_type(8)));

// 2D TDM: tile [tile_d0 x tile_d1] elements (f32), row stride = stride0 elems,
// landing contiguously at LDS byte offset lds_off. Issued per-wave; EXEC ignored.
__device__ __forceinline__ void tdm_load_to_lds_2d(unsigned lds_off, const void* gptr,
                                                   unsigned tensor_d0, unsigned tensor_d1,
                                                   unsigned tile_d0, unsigned tile_d1,
                                                   unsigned long long stride0_elems) {
  unsigned long long ga = (unsigned long long)(uintptr_t)gptr;
  u32x4 g0;
  g0[0] = 1u;                                                 // count=1, user descriptor
  g0[1] = lds_off;                                            // lds_addr (bytes)
  g0[2] = (unsigned)(ga & 0xFFFFFFFFull);                     // global_addr[31:0]
  g0[3] = (unsigned)((ga >> 32) & 0x1FFFFFFull) | (2u << 30); // global_addr[56:32] | type=2
  i32x8 g1;
  g1[0] = (int)(2u << 16);                                    // wg_mask=0, data_size=4B
  g1[1] = (int)((tensor_d0 & 0xFFFFu) << 16);                 // abar=0 | tensor_dim0[15:0]
  g1[2] = (int)(((tensor_d0 >> 16) & 0xFFFFu) | ((tensor_d1 & 0xFFFFu) << 16));
  g1[3] = (int)(((tensor_d1 >> 16) & 0xFFFFu) | ((tile_d0 & 0xFFFFu) << 16));
  g1[4] = (int)(tile_d1 & 0xFFFFu);                           // tile_dim1 | tile_dim2=0
  g1[5] = (int)(unsigned)(stride0_elems & 0xFFFFFFFFull);     // tensor_dim0_stride[31:0]
  g1[6] = (int)(unsigned)((stride0_elems >> 32) & 0xFFFFull); // stride0[47:32] | stride1 lo=0
  g1[7] = 0;
  i32x4 gz = (i32x4)0;
#if defined(__clang_major__) && (__clang_major__ >= 23)
  i32x8 gz8 = (i32x8)0;
  __builtin_amdgcn_tensor_load_to_lds(g0, g1, gz, gz, gz8, 0);
#else
  __builtin_amdgcn_tensor_load_to_lds(g0, g1, gz, gz, 0);
#endif
}
#endif // HAS_TDM

// ---------------- kernel 1: per-(image,class) top-500 + NMS ----------------
// grid = (NCLS, BATCH), 256 threads. Dynamic LDS = 180256 bytes.
// Phase A double-buffers TDM column tiles: DMA of tile t+1 overlaps filtering of tile t.

__global__ void __launch_bounds__(256)
fd_perclass(const float* __restrict__ boxes, const float* __restrict__ cls,
            float* __restrict__ cand_scores, float* __restrict__ cand_boxes) {
  const int c   = blockIdx.x;
  const int b   = blockIdx.y;
  const int tid = threadIdx.x;
  const int nt  = blockDim.x;

  extern __shared__ unsigned char smem[];
  unsigned long long* sortKey = (unsigned long long*)smem;            // 512 * 8      @0
  float*    tileBuf0  = (float*)(smem + 4096);                        // TILE         @4096
  float*    tileBuf1  = (float*)(smem + 4096 + TILE * 4);             // TILE         @20480
  unsigned* candKey   = (unsigned*)(smem + 4096 + 2 * TILE * 4);      // MAXC         @36864
  unsigned* candIdx   = candKey + MAXC;                               // MAXC         @102400
  float*    nmsBox    = (float*)(candIdx + MAXC);                     // 512*4        @167936
  float*    nmsArea   = nmsBox + 512 * 4;                             // 512          @176128
  unsigned* hist      = (unsigned*)(nmsArea + 512);                   // 256          @178176
  unsigned* shv       = hist + 256;                                   // 8 scalars    @179200
  unsigned char* supp = (unsigned char*)(shv + 8);                    // 512          @179232
  unsigned char* kept = supp + 512;                                   // 512          @179744

  if (tid == 0) shv[0] = 0u;
  __syncthreads();

  const float* colBase = cls + ((size_t)b * NBOX * NCLS + (size_t)c);

  // Phase A: stream the class column through LDS tiles, collect candidates > 0.05
#if HAS_TDM
  if ((tid >> 5) == 0) {                 // wave 0 primes the pipeline with tile 0
    tdm_load_to_lds_2d((unsigned)(uintptr_t)tileBuf0, colBase,
                       1u, (unsigned)((NBOX < TILE) ? NBOX : TILE),
                       1u, (unsigned)TILE, (unsigned long long)NCLS);
  }
  for (int t = 0; t < NTILES; ++t) {
    const int t0 = t * TILE;
    int rows = NBOX - t0; if (rows > TILE) rows = TILE;
    const float* buf = (t & 1) ? tileBuf1 : tileBuf0;
    if ((tid >> 5) == 0) {
      if (t + 1 < NTILES) {              // prefetch tile t+1 into the other buffer
        const int t1 = t0 + TILE;
        int rows1 = NBOX - t1; if (rows1 > TILE) rows1 = TILE;
        float* nbuf = (t & 1) ? tileBuf0 : tileBuf1;
        tdm_load_to_lds_2d((unsigned)(uintptr_t)nbuf, colBase + (size_t)t1 * NCLS,
                           1u, (unsigned)rows1, 1u, (unsigned)TILE,
                           (unsigned long long)NCLS);
        __builtin_amdgcn_s_wait_tensorcnt(1);   // in-order: tile t complete
      } else {
        __builtin_amdgcn_s_wait_tensorcnt(0);   // drain last tile
      }
    }
    __syncthreads();                     // tile t visible to the whole WG
    for (int r = tid; r < rows; r += nt) {
      float s = buf[r];
      if (s > SCORE_T) {
        unsigned pos = atomicAdd(&shv[0], 1u);
        if (pos < MAXC) { candKey[pos] = __float_as_uint(s); candIdx[pos] = (unsigned)(t0 + r); }
      }
    }
    __syncthreads();                     // done reading buf before it's reused at t+2
  }
#else
  for (int t0 = 0; t0 < NBOX; t0 += TILE) {
    int rows = NBOX - t0; if (rows > TILE) rows = TILE;
    for (int r = tid; r < TILE; r += nt)
      tileBuf0[r] = (r < rows) ? colBase[(size_t)(t0 + r) * NCLS] : 0.0f;
    __syncthreads();
    for (int r = tid; r < rows; r += nt) {
      float s = tileBuf0[r];
      if (s > SCORE_T) {
        unsigned pos = atomicAdd(&shv[0], 1u);
        if (pos < MAXC) { candKey[pos] = __float_as_uint(s); candIdx[pos] = (unsigned)(t0 + r); }
      }
    }
    __syncthreads();
  }
#endif

  unsigned M = shv[0]; if (M > MAXC) M = MAXC;

  // Phase B: exact top-KCAND via 64-bit radix select on unique composite keys
  unsigned long long thr = 0ull;
  if (M > KCAND) {
    unsigned long long prefix = 0ull;
    unsigned remaining = KCAND;
    for (int shift = 56; shift >= 0; shift -= 8) {
      for (int i = tid; i < 256; i += nt) hist[i] = 0u;
      __syncthreads();
      unsigned long long hiMask = (shift == 56) ? 0ull : ((~0ull) << (unsigned)(shift + 8));
      for (unsigned i = (unsigned)tid; i < M; i += (unsigned)nt) {
        unsigned long long key = key64_of(candKey[i], candIdx[i]);
        if ((key & hiMask) == prefix)
          atomicAdd(&hist[(unsigned)((key >> shift) & 0xFFull)], 1u);
      }
      __syncthreads();
      if (tid == 0) {
        unsigned cum = 0; int bsel = 0;
        for (int bb = 255; bb >= 0; --bb) {
          unsigned cc = hist[bb];
          if (cum + cc >= remaining || bb == 0) { bsel = bb; break; }
          cum += cc;
        }
        shv[2] = (unsigned)bsel; shv[3] = cum;
      }
      __syncthreads();
      prefix |= ((unsigned long long)shv[2]) << shift;
      remaining -= shv[3];
      __syncthreads();
    }
    thr = prefix;      // exact 500th-largest key
  }

  if (tid == 0) shv[1] = 0u;
  __syncthreads();
  for (unsigned i = (unsigned)tid; i < M; i += (unsigned)nt) {
    unsigned long long key = key64_of(candKey[i], candIdx[i]);
    if (M <= KCAND || key >= thr) {
      unsigned pos = atomicAdd(&shv[1], 1u);
      if (pos < 512u) sortKey[pos] = key;
    }
  }
  __syncthreads();
  unsigned cnt = shv[1]; if (cnt > KCAND) cnt = KCAND;
  for (int i = tid; i < 512; i += nt) if ((unsigned)i >= cnt) sortKey[i] = 0ull;
  __syncthreads();
  bitonic_desc_512(sortKey, tid, nt);
  const int selCnt = (int)cnt;

  // Phase C: gather boxes for sorted candidates, greedy NMS
  for (int k = tid; k < 512; k += nt) {
    float y1 = 0.f, x1 = 0.f, y2 = 0.f, x2 = 0.f;
    if (k < selCnt) {
      unsigned idx = ~(unsigned)(sortKey[k] & 0xFFFFFFFFull);
      const float* bp = boxes + ((size_t)b * NBOX + idx) * 4;
      y1 = bp[0]; x1 = bp[1]; y2 = bp[2]; x2 = bp[3];
    }
    nmsBox[k * 4 + 0] = y1; nmsBox[k * 4 + 1] = x1;
    nmsBox[k * 4 + 2] = y2; nmsBox[k * 4 + 3] = x2;
    float h = y2 - y1; if (h < 0.f) h = 0.f;
    float w = x2 - x1; if (w < 0.f) w = 0.f;
    nmsArea[k] = h * w;
    supp[k] = 0; kept[k] = 0;
  }
  __syncthreads();

  for (int i = 0; i < KCAND; ++i) {
    if (tid == 0) {
      unsigned kp = (i < selCnt && !supp[i]) ? 1u : 0u;
      kept[i] = (unsigned char)kp;
      shv[4] = kp;
    }
    __syncthreads();
    if (shv[4]) {
      const float by1 = nmsBox[i * 4 + 0], bx1 = nmsBox[i * 4 + 1];
      const float by2 = nmsBox[i * 4 + 2], bx2 = nmsBox[i * 4 + 3];
      const float ba = nmsArea[i];
      for (int j = i + 1 + tid; j < selCnt; j += nt) {
        float yy1 = fmaxf(by1, nmsBox[j * 4 + 0]);
        float xx1 = fmaxf(bx1, nmsBox[j * 4 + 1]);
        float yy2 = fminf(by2, nmsBox[j * 4 + 2]);
        float xx2 = fminf(bx2, nmsBox[j * 4 + 3]);
        float ih = yy2 - yy1; if (ih < 0.f) ih = 0.f;
        float iw = xx2 - xx1; if (iw < 0.f) iw = 0.f;
        float inter = ih * iw;
        float iou = inter / (ba + nmsArea[j] - inter + 1e-9f);
        if (iou > NMS_TH) supp[j] = 1;
      }
    }
    __syncthreads();
  }

  if (tid == 0) {       // cap at MAX_DETECTIONS per class (cumsum rule)
    int cum = 0;
    for (int i = 0; i < KCAND; ++i)
      if (kept[i]) { ++cum; if (cum > MAXDET) kept[i] = 0; }
  }
  __syncthreads();

  const float NEG_INF = -__builtin_inff();
  const size_t base = ((size_t)b * NCLS + (size_t)c) * KCAND;
  for (int k = tid; k < KCAND; k += nt) {
    float scv = NEG_INF;
    if (k < selCnt && kept[k]) scv = from_ord((unsigned)(sortKey[k] >> 32));
    cand_scores[base + k] = scv;
    float* ob = cand_boxes + (base + k) * 4;
    ob[0] = nmsBox[k * 4 + 0]; ob[1] = nmsBox[k * 4 + 1];
    ob[2] = nmsBox[k * 4 + 2]; ob[3] = nmsBox[k * 4 + 3];
  }
}

// ---------------- kernel 2: per-image global top-300 ----------------
// grid = (BATCH), 512 threads. Dynamic LDS = 165152 bytes.

__global__ void __launch_bounds__(512)
fd_final(const float* __restrict__ cand_scores, const float* __restrict__ cand_boxes,
         float* __restrict__ out) {
  const int b   = blockIdx.x;
  const int tid = threadIdx.x;
  const int nt  = blockDim.x;
  const int M2  = NCLS * KCAND;   // 40000

  extern __shared__ unsigned char smem[];
  unsigned long long* sortKey = (unsigned long long*)smem;             // 512 * 8
  float*    sc   = (float*)(smem + 4096);                              // 40000
  unsigned* hist = (unsigned*)(smem + 4096 + (size_t)M2 * 4);          // 256
  unsigned* shv  = hist + 256;                                         // 8

  const float* src = cand_scores + (size_t)b * M2;
#if HAS_TDM
  if ((tid >> 5) == 0) {          // one 160 KB contiguous TDM transfer into LDS
    tdm_load_to_lds_2d((unsigned)(uintptr_t)sc, src,
                       (unsigned)M2, 1u, (unsigned)M2, 1u,
                       (unsigned long long)M2);
    __builtin_amdgcn_s_wait_tensorcnt(0);
  }
  __syncthreads();
#else
  for (int i = tid; i < M2; i += nt) sc[i] = src[i];
  __syncthreads();
#endif

  unsigned long long prefix = 0ull;
  unsigned remaining = MAXDET;
  for (int shift = 56; shift >= 0; shift -= 8) {
    for (int i = tid; i < 256; i += nt) hist[i] = 0u;
    __syncthreads();
    unsigned long long hiMask = (shift == 56) ? 0ull : ((~0ull) << (unsigned)(shift + 8));
    for (int i = tid; i < M2; i += nt) {
      unsigned long long key = key64_of(__float_as_uint(sc[i]), (unsigned)i);
      if ((key & hiMask) == prefix)
        atomicAdd(&hist[(unsigned)((key >> shift) & 0xFFull)], 1u);
    }
    __syncthreads();
    if (tid == 0) {
      unsigned cum = 0; int bsel = 0;
      for (int bb = 255; bb >= 0; --bb) {
        unsigned cc = hist[bb];
        if (cum + cc >= remaining || bb == 0) { bsel = bb; break; }
        cum += cc;
      }
      shv[2] = (unsigned)bsel; shv[3] = cum;
    }
    __syncthreads();
    prefix |= ((unsigned long long)shv[2]) << shift;
    remaining -= shv[3];
    __syncthreads();
  }
  const unsigned long long thr = prefix;

  if (tid == 0) shv[1] = 0u;
  __syncthreads();
  for (int i = tid; i < M2; i += nt) {
    unsigned long long key = key64_of(__float_as_uint(sc[i]), (unsigned)i);
    if (key >= thr) {
      unsigned pos = atomicAdd(&shv[1], 1u);
      if (pos < 512u) sortKey[pos] = key;
    }
  }
  __syncthreads();
  unsigned cnt = shv[1]; if (cnt > MAXDET) cnt = MAXDET;
  for (int i = tid; i < 512; i += nt) if ((unsigned)i >= cnt) sortKey[i] = 0ull;
  __syncthreads();
  bitonic_desc_512(sortKey, tid, nt);

  const float NEG_INF = -__builtin_inff();
  float* outBoxes  = out;
  float* outScores = out + (size_t)BATCH * MAXDET * 4;
  float* outLabels = outScores + (size_t)BATCH * MAXDET;
  for (int k = tid; k < MAXDET; k += nt) {
    float b0 = -1.f, b1 = -1.f, b2 = -1.f, b3 = -1.f, scv = -1.f, lab = -1.f;
    if ((unsigned)k < cnt) {
      unsigned long long key = sortKey[k];
      unsigned flat = ~(unsigned)(key & 0xFFFFFFFFull);
      if (flat < (unsigned)M2) {
        float s = sc[flat];
        if (s != NEG_INF) {     // pad slots -> -1, matching reference
          int cid = (int)(flat / KCAND);
          const float* bp = cand_boxes + ((size_t)b * M2 + flat) * 4;
          b0 = bp[0]; b1 = bp[1]; b2 = bp[2]; b3 = bp[3];
          scv = s; lab = (float)cid;
        }
      }
    }
    const size_t o = (size_t)b * MAXDET + k;
    outBoxes[o * 4 + 0] = b0; outBoxes[o * 4 + 1] = b1;
    outBoxes[o * 4 + 2] = b2; outBoxes[o * 4 + 3] = b3;
    outScores[o] = scv;
    outLabels[o] = lab;
  }
}

// ---------------- host ----------------

extern "C" void kernel_launch(void* const* d_in, const int* in_sizes, int n_in,
                              void* d_out, int out_size, void* d_ws, size_t ws_size,
                              hipStream_t stream) {
  (void)in_sizes; (void)n_in; (void)out_size; (void)ws_size;
  const float* boxes = (const float*)d_in[0];
  const float* cls   = (const float*)d_in[1];
  float* out = (float*)d_out;

  float* cand_scores = (float*)d_ws;                                      // B*C*K floats
  float* cand_boxes  = cand_scores + (size_t)BATCH * NCLS * KCAND;        // B*C*K*4 floats

  const int LDS1 = 180256;   // see carve in fd_perclass (double-buffered tiles)
  const int LDS2 = 165152;   // see carve in fd_final
  (void)hipFuncSetAttribute(reinterpret_cast<const void*>(fd_perclass),
                            hipFuncAttributeMaxDynamicSharedMemorySize, LDS1);
  (void)hipFuncSetAttribute(reinterpret_cast<const void*>(fd_final),
                            hipFuncAttributeMaxDynamicSharedMemorySize, LDS2);

  fd_perclass<<<dim3(NCLS, BATCH), 256, LDS1, stream>>>(boxes, cls, cand_scores, cand_boxes);
  fd_final<<<dim3(BATCH), 512, LDS2, stream>>>(cand_scores, cand_boxes, out);
}